// AdaptiveAnchorConvolution_82806969466908
// MI455X (gfx1250) — compile-verified
//
#include <hip/hip_runtime.h>
#include <math.h>

#define N_ROWS  131072
#define FEAT    256
#define N_ANC   64
#define ANC     128
#define B3      1024           // blocks for weighted row-sum pass
#define RPB     (N_ROWS / B3)  // rows per block in that pass

typedef float v2f __attribute__((ext_vector_type(2)));
typedef float v8f __attribute__((ext_vector_type(8)));

// ---------- wave32 helpers ----------
__device__ __forceinline__ float waveSum(float x) {
    #pragma unroll
    for (int o = 16; o > 0; o >>= 1) x += __shfl_xor(x, o, 32);
    return x;
}
__device__ __forceinline__ float waveMax(float x) {
    #pragma unroll
    for (int o = 16; o > 0; o >>= 1) x = fmaxf(x, __shfl_xor(x, o, 32));
    return x;
}

// ---------- 1xK vector times KxN matrix via V_WMMA_F32_16X16X4_F32 ----------
// vec sits in LDS; only row m==0 of the A tile is populated (rest zero).
// A layout (16x4 f32): lane m=L%16, hi=L>=16; VGPR0 holds K=hi?2:0, VGPR1 K+1.
// B layout (4x16 f32): lane n=L%16; VGPR0 holds K=hi?2:0, VGPR1 K+1.
// D row 0 = acc[0] on lanes 0..15 (N = lane).
__device__ __forceinline__ void vecmat_wmma(const float* vec, const float* mat,
                                            int ld, int trans, int K, int N,
                                            float* out, int lane) {
    const int m  = lane & 15;
    const int hi = (lane >> 4) & 1;
    for (int nt = 0; nt < (N >> 4); ++nt) {
        v8f acc = {};
        const int n = (nt << 4) + m;
        for (int k0 = 0; k0 < K; k0 += 4) {
            const int ka0 = k0 + (hi ? 2 : 0);
            v2f A, B;
            A.x = (m == 0) ? vec[ka0]     : 0.0f;
            A.y = (m == 0) ? vec[ka0 + 1] : 0.0f;
            if (trans) { B.x = mat[n * ld + ka0];       B.y = mat[n * ld + ka0 + 1]; }
            else       { B.x = mat[ka0 * ld + n];       B.y = mat[(ka0 + 1) * ld + n]; }
            acc = __builtin_amdgcn_wmma_f32_16x16x4_f32(
                false, A, false, B, (short)0, acc, false, false);
        }
        if (lane < 16) out[(nt << 4) + lane] = acc[0];
    }
}

// ---------- kernel 0: v = W_send @ a_send[ANC:]  (256-vector) ----------
__global__ void k0_send_vec(const float* __restrict__ Wsend,
                            const float* __restrict__ a_send,
                            float* __restrict__ v) {
    int k = threadIdx.x; // 256
    float acc = 0.f;
    #pragma unroll 4
    for (int d = 0; d < ANC; ++d) acc += Wsend[k * ANC + d] * a_send[ANC + d];
    v[k] = acc;
}

// ---------- kernel 1: per-row LN stats + score s_i = nf_i . v ----------
// 256 threads = 8 waves, one row per wave (row = 1 KB, lane owns 8 floats).
__global__ void k1_rowstats(const float* __restrict__ feat,
                            const float* __restrict__ g, const float* __restrict__ b,
                            const float* __restrict__ v,
                            float* __restrict__ sArr, float* __restrict__ muArr,
                            float* __restrict__ rsArr) {
    const int lane = threadIdx.x & 31;
    const int wid  = threadIdx.x >> 5;
    const int row  = blockIdx.x * 8 + wid;
    const int c    = lane * 8;
    const float4* fp = (const float4*)(feat + (size_t)row * FEAT + c);
    float4 x0 = fp[0], x1 = fp[1];
    float xs[8] = {x0.x, x0.y, x0.z, x0.w, x1.x, x1.y, x1.z, x1.w};

    float sum = 0.f, sq = 0.f;
    #pragma unroll
    for (int j = 0; j < 8; ++j) { sum += xs[j]; sq += xs[j] * xs[j]; }
    sum = waveSum(sum); sq = waveSum(sq);
    const float mu  = sum * (1.0f / FEAT);
    const float var = sq * (1.0f / FEAT) - mu * mu;
    const float rs  = rsqrtf(var + 1e-5f);

    const float4* gp = (const float4*)(g + c);
    const float4* bp = (const float4*)(b + c);
    const float4* vp = (const float4*)(v + c);
    float4 g0 = gp[0], g1 = gp[1], b0 = bp[0], b1 = bp[1], v0 = vp[0], v1 = vp[1];
    float gs[8] = {g0.x, g0.y, g0.z, g0.w, g1.x, g1.y, g1.z, g1.w};
    float bs[8] = {b0.x, b0.y, b0.z, b0.w, b1.x, b1.y, b1.z, b1.w};
    float vs[8] = {v0.x, v0.y, v0.z, v0.w, v1.x, v1.y, v1.z, v1.w};

    float sp = 0.f;
    #pragma unroll
    for (int j = 0; j < 8; ++j)
        sp += ((xs[j] - mu) * rs * gs[j] + bs[j]) * vs[j];
    sp = waveSum(sp);

    if (lane == 0) { sArr[row] = sp; muArr[row] = mu; rsArr[row] = rs; }
}

// ---------- kernel 2: global softmax stats (max M, denom L) ----------
__global__ void k2_softmax_stats(const float* __restrict__ s, float* __restrict__ ML) {
    __shared__ float red[32];
    const int t = threadIdx.x; // 1024
    float m = -INFINITY;
    for (int i = t; i < N_ROWS; i += 1024) m = fmaxf(m, s[i]);
    m = waveMax(m);
    if ((t & 31) == 0) red[t >> 5] = m;
    __syncthreads();
    if (t < 32) { float x = red[t]; x = waveMax(x); if (t == 0) red[0] = x; }
    __syncthreads();
    const float M = red[0];
    __syncthreads();
    float sum = 0.f;
    for (int i = t; i < N_ROWS; i += 1024) sum += expf(s[i] - M);
    sum = waveSum(sum);
    if ((t & 31) == 0) red[t >> 5] = sum;
    __syncthreads();
    if (t == 0) {
        float L = 0.f;
        for (int i = 0; i < 32; ++i) L += red[i];   // fixed order: deterministic
        ML[0] = M; ML[1] = L;
    }
}

// ---------- kernel 3: partial weighted row-sum  zpart[blk][t] ----------
// thread t owns column t; each block accumulates RPB rows serially (deterministic).
__global__ void k3_zpart(const float* __restrict__ feat,
                         const float* __restrict__ sArr, const float* __restrict__ muArr,
                         const float* __restrict__ rsArr, const float* __restrict__ ML,
                         float* __restrict__ zpart) {
    const int t = threadIdx.x; // 256
    const int row0 = blockIdx.x * RPB;
    const float M = ML[0];
    const float invL = 1.0f / ML[1];
    float acc = 0.f;
    for (int r = 0; r < RPB; ++r) {
        const int row = row0 + r;
        const float w = expf(sArr[row] - M) * invL;
        const float x = feat[(size_t)row * FEAT + t];
        acc += w * ((x - muArr[row]) * rsArr[row]);
    }
    zpart[blockIdx.x * FEAT + t] = acc;
}

// ---------- kernel 4: tiny dense chain (WMMA) -> sin(h2) ----------
__global__ void k4_tiny(const float* __restrict__ zpart,
                        const float* __restrict__ g_feat, const float* __restrict__ b_feat,
                        const float* __restrict__ Wsend, const float* __restrict__ anchors,
                        const float* __restrict__ g_anc, const float* __restrict__ b_anc,
                        const float* __restrict__ Wrecv, float* __restrict__ sinh2) {
    __shared__ float zs[FEAT], pooled[ANC], proj[N_ANC], na[N_ANC], h2[FEAT];
    const int t = threadIdx.x; // 256
    float acc = 0.f;
    for (int i = 0; i < B3; ++i) acc += zpart[i * FEAT + t];  // fixed order
    zs[t] = g_feat[t] * acc + b_feat[t];
    __syncthreads();

    if (t < 32) {  // one full wave, EXEC all-ones for WMMA
        const int lane = t;
        // pooled[128] = z(1x256) @ W_send(256x128)
        vecmat_wmma(zs, Wsend, ANC, 0, FEAT, ANC, pooled, lane);
        // proj[64] = pooled(1x128) @ anchors^T(128x64); anchors is [64,128] row-major
        vecmat_wmma(pooled, anchors, ANC, 1, ANC, N_ANC, proj, lane);
        // LayerNorm over 64
        float a0 = proj[lane], a1 = proj[lane + 32];
        float sum = waveSum(a0 + a1);
        float sq  = waveSum(a0 * a0 + a1 * a1);
        const float mu  = sum * (1.0f / N_ANC);
        const float var = sq * (1.0f / N_ANC) - mu * mu;
        const float rs  = rsqrtf(var + 1e-5f);
        na[lane]      = (a0 - mu) * rs * g_anc[lane]      + b_anc[lane];
        na[lane + 32] = (a1 - mu) * rs * g_anc[lane + 32] + b_anc[lane + 32];
        // h2[256] = na(1x64) @ W_recv(64x256)
        vecmat_wmma(na, Wrecv, FEAT, 0, N_ANC, FEAT, h2, lane);
    }
    __syncthreads();
    sinh2[t] = sinf(h2[t]);
}

// ---------- kernel 5: out = features + broadcast(sin(h2)) ----------
__global__ void k5_out(const float* __restrict__ feat, const float* __restrict__ sinh2,
                       float* __restrict__ out, int total4) {
    int i = blockIdx.x * blockDim.x + threadIdx.x;
    const int stride = gridDim.x * blockDim.x;
    for (; i < total4; i += stride) {
        const int c4 = (i & 63) * 4;           // FEAT/4 = 64 float4 per row
        float4 f = ((const float4*)feat)[i];
        float4 s = *(const float4*)(sinh2 + c4);
        float4 o = {f.x + s.x, f.y + s.y, f.z + s.z, f.w + s.w};
        ((float4*)out)[i] = o;
    }
}

extern "C" void kernel_launch(void* const* d_in, const int* in_sizes, int n_in,
                              void* d_out, int out_size, void* d_ws, size_t ws_size,
                              hipStream_t stream) {
    const float* features = (const float*)d_in[0];  // [131072,256]
    const float* W_send   = (const float*)d_in[1];  // [256,128]
    const float* a_send   = (const float*)d_in[2];  // [256,1]
    const float* W_recv   = (const float*)d_in[3];  // [64,256]
    // d_in[4] = a_recv: provably dead (second softmax is uniform)
    const float* anchors  = (const float*)d_in[5];  // [64,128]
    const float* g_feat   = (const float*)d_in[6];
    const float* b_feat   = (const float*)d_in[7];
    const float* g_anc    = (const float*)d_in[8];
    const float* b_anc    = (const float*)d_in[9];
    float* out = (float*)d_out;

    float* w = (float*)d_ws;
    float* v_vec  = w;                          // 256
    float* sArr   = v_vec + 256;                // N
    float* muArr  = sArr + N_ROWS;              // N
    float* rsArr  = muArr + N_ROWS;             // N
    float* ML     = rsArr + N_ROWS;             // 2 (padded to 8)
    float* sinh2  = ML + 8;                     // 256
    float* zpart  = sinh2 + 256;                // B3*256
    (void)ws_size; (void)in_sizes; (void)n_in; (void)out_size;

    k0_send_vec<<<1, 256, 0, stream>>>(W_send, a_send, v_vec);
    k1_rowstats<<<N_ROWS / 8, 256, 0, stream>>>(features, g_feat, b_feat, v_vec,
                                                sArr, muArr, rsArr);
    k2_softmax_stats<<<1, 1024, 0, stream>>>(sArr, ML);
    k3_zpart<<<B3, 256, 0, stream>>>(features, sArr, muArr, rsArr, ML, zpart);
    k4_tiny<<<1, 256, 0, stream>>>(zpart, g_feat, b_feat, W_send, anchors,
                                   g_anc, b_anc, W_recv, sinh2);
    const int total4 = N_ROWS * FEAT / 4;
    k5_out<<<8192, 256, 0, stream>>>(features, sinh2, out, total4);
}